// NoisyTopKRouter_81484119540362
// MI455X (gfx1250) — compile-verified
//
#include <hip/hip_runtime.h>
#include <hip/hip_bf16.h>

// MoE top-2 router for MI455X (gfx1250, wave32).
// logits = x[N,2048] @ W[16,2048]^T via V_WMMA_F32_16X16X4_F32, then per-token
// top-2 + softmax. Bandwidth-bound: x read = 1.07 GB -> ~46us floor @ 23.3TB/s.

typedef float v2f __attribute__((ext_vector_type(2)));
typedef float v8f __attribute__((ext_vector_type(8)));

static constexpr int kD = 2048;            // feature dim
static constexpr int kE = 16;              // experts
static constexpr int kTopK = 2;
static constexpr int kTokPerWave = 16;     // one WMMA M-tile per wave
static constexpr int kWavesPerBlk = 8;     // 256 threads
static constexpr int kTokPerBlk = kTokPerWave * kWavesPerBlk;  // 128
static constexpr int kKChunk = 32;         // 8 WMMA K-tiles per iteration

__global__ void __launch_bounds__(256)
moe_router_topk(const float* __restrict__ x, const float* __restrict__ W,
                long long* __restrict__ idx_out, float* __restrict__ gate_out)
{
    __shared__ float lds_logits[kTokPerBlk * kE];

    const int tid  = threadIdx.x;
    const int lane = tid & 31;
    const int wave = tid >> 5;
    const int half = lane >> 4;     // 0: K={0,1}, 1: K={2,3} (f32 A/B layout)
    const int l15  = lane & 15;     // A: row M; B: col N
    const int kk   = half * 2;

    const long long blockTokBase = (long long)blockIdx.x * kTokPerBlk;
    const long long waveTokBase  = blockTokBase + (long long)wave * kTokPerWave;

    // Per-lane base pointers: A lane reads row (token), B lane reads W row (expert).
    const float* __restrict__ aPtr = x + (size_t)(waveTokBase + l15) * kD + kk;
    const float* __restrict__ bPtr = W + (size_t)l15 * kD + kk;

    v8f c = {0.f, 0.f, 0.f, 0.f, 0.f, 0.f, 0.f, 0.f};

    for (int k0 = 0; k0 < kD; k0 += kKChunk) {
        v2f a[8], b[8];
#pragma unroll
        for (int t = 0; t < 8; ++t) {
            a[t] = *(const v2f*)(aPtr + k0 + 4 * t);   // global_load_b64
            b[t] = *(const v2f*)(bPtr + k0 + 4 * t);   // hits L1/L2 (W = 128KB)
        }
#pragma unroll
        for (int t = 0; t < 8; ++t) {
            // D = A(16x4) * B(4x16) + C ; args: (neg_a, A, neg_b, B, c_mod, C, reuse_a, reuse_b)
            c = __builtin_amdgcn_wmma_f32_16x16x4_f32(
                    false, a[t], false, b[t], (short)0, c, false, false);
        }
    }

    // C layout: VGPR r, lanes 0-15 -> (M=r, N=lane); lanes 16-31 -> (M=r+8, N=lane-16).
    // Scatter logits to LDS as [token][expert].
#pragma unroll
    for (int r = 0; r < 8; ++r) {
        int tok = wave * kTokPerWave + half * 8 + r;
        lds_logits[tok * kE + l15] = c[r];
    }
    __syncthreads();

    // 128 threads: one token each -> top-2 over 16 logits + 2-way softmax.
    if (tid < kTokPerBlk) {
        const float* lg = &lds_logits[tid * kE];
        float b0 = lg[0];
        int   i0 = 0;
#pragma unroll
        for (int e = 1; e < kE; ++e) {
            float v = lg[e];
            if (v > b0) { b0 = v; i0 = e; }   // strict '>' == top_k lowest-index tie rule
        }
        float b1 = -__builtin_inff();
        int   i1 = 0;
#pragma unroll
        for (int e = 0; e < kE; ++e) {
            if (e != i0) {
                float v = lg[e];
                if (v > b1) { b1 = v; i1 = e; }
            }
        }
        // softmax([b0, b1]) with b0 >= b1 (numerically stable form)
        float ex  = __expf(b1 - b0);
        float inv = 1.0f / (1.0f + ex);
        long long g = blockTokBase + tid;
        idx_out[2 * g]     = (long long)i0;
        idx_out[2 * g + 1] = (long long)i1;
        gate_out[2 * g]     = inv;
        gate_out[2 * g + 1] = ex * inv;
    }
}

extern "C" void kernel_launch(void* const* d_in, const int* in_sizes, int n_in,
                              void* d_out, int out_size, void* d_ws, size_t ws_size,
                              hipStream_t stream) {
    const float* x = (const float*)d_in[0];   // [N, 2048] fp32
    const float* W = (const float*)d_in[1];   // [16, 2048] fp32
    const int N = in_sizes[0] / kD;

    // Outputs concatenated in return order: int64 idx [N,2], then fp32 gates [N,2].
    long long* idx_out  = (long long*)d_out;
    float*     gate_out = (float*)((char*)d_out + (size_t)N * kTopK * sizeof(long long));

    const int blocks = (N + kTokPerBlk - 1) / kTokPerBlk;
    moe_router_topk<<<blocks, 256, 0, stream>>>(x, W, idx_out, gate_out);
}